// SoftAUAvULoss_6081673691430
// MI455X (gfx1250) — compile-verified
//
#include <hip/hip_runtime.h>
#include <hip/hip_bf16.h>

typedef float v2f __attribute__((ext_vector_type(2)));
typedef float v4f __attribute__((ext_vector_type(4)));
typedef float v8f __attribute__((ext_vector_type(8)));

#define NCLS 1000
#define EPS10 1e-10f
#define EPS8  1e-8f
// log(1e-8) so we only pay one v_log per element:
// log(max(p,1e-8)) == max(log(max(p,1e-10)), log(1e-8))
#define LOG_EPS8 (-18.420680743952367f)
#define ENTMAX 6.907755278982137f   // log(1000)

// ---------------------------------------------------------------------------
// Full-wave32 sum via V_WMMA_F32_16X16X4_F32.
// A layout (32-bit A 16x4): VGPR0 lanes0-15 -> K=0 (M=lane), lanes16-31 -> K=2
// (M=lane-16); VGPR1 -> K=1/K=3.  With a={x,0}, B=ones:
//   D(m,n) = x_m + x_{m+16}  (all n)
// D layout: vgpr r, lanes0-15 -> M=r, lanes16-31 -> M=r+8.  So sum of d[0..7]
// on lanes<16 gives S_0..S_7, on lanes>=16 gives S_8..S_15; one xor-16 shuffle
// completes the full 32-lane sum.  EXEC must be all ones at the call site.
// ---------------------------------------------------------------------------
__device__ __forceinline__ float wave_sum_wmma(float x) {
    v2f a; a[0] = x;    a[1] = 0.0f;
    v2f b; b[0] = 1.0f; b[1] = 1.0f;
    v8f c = {};
    v8f d = __builtin_amdgcn_wmma_f32_16x16x4_f32(false, a, false, b,
                                                  (short)0, c, false, false);
    float s = ((d[0] + d[1]) + (d[2] + d[3])) + ((d[4] + d[5]) + (d[6] + d[7]));
    s += __shfl_xor(s, 16, 32);
    return s;
}

__device__ __forceinline__ float wave_min32(float v) {
    #pragma unroll
    for (int o = 16; o > 0; o >>= 1) v = fminf(v, __shfl_xor(v, o, 32));
    return v;
}
__device__ __forceinline__ float wave_max32(float v) {
    #pragma unroll
    for (int o = 16; o > 0; o >>= 1) v = fmaxf(v, __shfl_xor(v, o, 32));
    return v;
}

// ---------------------------------------------------------------------------
// Pass 1: one wave per row.  Streams probs/y/weights once with NT (streaming)
// b128 loads, fuses entropy, CE, focal, argmax(probs), argmax(y) (row 0).
// ---------------------------------------------------------------------------
__device__ __forceinline__ void elem_accum(float p, float yv, float wv, int j,
                                           float& ent, float& ce, float& foc,
                                           float& bv, int& bi,
                                           float& ybv, int& ybi) {
    float l10 = __logf(fmaxf(p, EPS10));          // log(clip(p,1e-10))
    ent -= p * l10;
    float l8 = fmaxf(l10, LOG_EPS8);              // log(clip(p,1e-8))
    float base = yv * l8;
    ce  -= base;
    foc -= base * wv * (1.0f - p);
    if (p  > bv)  { bv  = p;  bi  = j; }
    if (yv > ybv) { ybv = yv; ybi = j; }
}

__device__ __forceinline__ v4f ld_nt4(const float* p) {
    return __builtin_nontemporal_load((const v4f*)p);   // global_load_b128 NT
}

__global__ __launch_bounds__(256)
void savu_pass1(const float* __restrict__ probs,
                const float* __restrict__ y,
                const float* __restrict__ w,
                float* __restrict__ unc,
                float* __restrict__ ce_row,
                float* __restrict__ foc_row,
                int*   __restrict__ pred,
                int*   __restrict__ label0,
                int n) {
    const int lane = threadIdx.x & 31;
    const int row  = blockIdx.x * 8 + (threadIdx.x >> 5);
    if (row >= n) return;                    // wave-uniform

    const size_t off = (size_t)row * NCLS;
    const float* pr = probs + off;
    const float* yr = y     + off;
    const float* wr = w     + off;

    float ent = 0.0f, ce = 0.0f, foc = 0.0f;
    float bv  = -INFINITY; int bi  = 0x7fffffff;
    float ybv = -INFINITY; int ybi = 0x7fffffff;

    // 1000 = 7*128 + 26*4 ; all b128 loads are 16B aligned (row stride 4000B)
    #pragma unroll
    for (int t = 0; t < 7; ++t) {
        int base = t * 128 + lane * 4;
        v4f p4 = ld_nt4(pr + base);
        v4f y4 = ld_nt4(yr + base);
        v4f w4 = ld_nt4(wr + base);
        #pragma unroll
        for (int c4 = 0; c4 < 4; ++c4)
            elem_accum(p4[c4], y4[c4], w4[c4], base + c4,
                       ent, ce, foc, bv, bi, ybv, ybi);
    }
    if (lane < 26) {
        int base = 896 + lane * 4;
        v4f p4 = ld_nt4(pr + base);
        v4f y4 = ld_nt4(yr + base);
        v4f w4 = ld_nt4(wr + base);
        #pragma unroll
        for (int c4 = 0; c4 < 4; ++c4)
            elem_accum(p4[c4], y4[c4], w4[c4], base + c4,
                       ent, ce, foc, bv, bi, ybv, ybi);
    }

    // wave reductions: sums via WMMA, argmax via shuffles (first-occurrence)
    ent = wave_sum_wmma(ent);
    ce  = wave_sum_wmma(ce);
    foc = wave_sum_wmma(foc);
    #pragma unroll
    for (int o = 16; o > 0; o >>= 1) {
        float ov = __shfl_xor(bv, o, 32);  int oi = __shfl_xor(bi, o, 32);
        if (ov > bv || (ov == bv && oi < bi)) { bv = ov; bi = oi; }
        float oy = __shfl_xor(ybv, o, 32); int oj = __shfl_xor(ybi, o, 32);
        if (oy > ybv || (oy == ybv && oj < ybi)) { ybv = oy; ybi = oj; }
    }

    if (lane == 0) {
        unc[row]     = ent;
        ce_row[row]  = ce;
        foc_row[row] = foc;
        pred[row]    = bi;
        if (row == 0) *label0 = ybi;   // jnp.argmax(y) flattened == row0 label
    }
}

// ---------------------------------------------------------------------------
// Pass 2: single 1024-thread workgroup finisher.
// ---------------------------------------------------------------------------
__device__ __forceinline__ float block_sum(float v, float* red, int wid, int lane) {
    float s = wave_sum_wmma(v);
    if (lane == 0) red[wid] = s;
    __syncthreads();
    if (wid == 0) {
        float t = red[lane];
        t = wave_sum_wmma(t);
        if (lane == 0) red[0] = t;
    }
    __syncthreads();
    float r = red[0];
    __syncthreads();
    return r;
}
__device__ __forceinline__ float block_min(float v, float* red, int wid, int lane) {
    v = wave_min32(v);
    if (lane == 0) red[wid] = v;
    __syncthreads();
    if (wid == 0) {
        float t = wave_min32(red[lane]);
        if (lane == 0) red[0] = t;
    }
    __syncthreads();
    float r = red[0];
    __syncthreads();
    return r;
}
__device__ __forceinline__ float block_max(float v, float* red, int wid, int lane) {
    v = wave_max32(v);
    if (lane == 0) red[wid] = v;
    __syncthreads();
    if (wid == 0) {
        float t = wave_max32(red[lane]);
        if (lane == 0) red[0] = t;
    }
    __syncthreads();
    float r = red[0];
    __syncthreads();
    return r;
}

__global__ __launch_bounds__(1024)
void savu_pass2(const float* __restrict__ unc,
                const float* __restrict__ ce_row,
                const float* __restrict__ foc_row,
                const int*   __restrict__ pred,
                const int*   __restrict__ label0,
                float* __restrict__ out,
                int n) {
    __shared__ float red[32];
    __shared__ float avu_s[21];

    const int tid  = threadIdx.x;
    const int wid  = tid >> 5;
    const int lane = tid & 31;

    // Phase A: umin/umax of unc, means of ce & focal
    float mn = INFINITY, mx = -INFINITY, ces = 0.0f, fos = 0.0f;
    for (int r = tid; r < n; r += 1024) {
        float u = unc[r];
        mn = fminf(mn, u); mx = fmaxf(mx, u);
        ces += ce_row[r];  fos += foc_row[r];
    }
    float ce_mean  = block_sum(ces, red, wid, lane) / (float)n;
    float foc_mean = block_sum(fos, red, wid, lane) / (float)n;
    float umin = block_min(mn, red, wid, lane);
    float umax = block_max(mx, red, wid, lane);
    float du   = umax - umin;
    int   lbl0 = *label0;

    // Phase B: 21 thresholds; per-threshold we only need
    //   num = n_ac + n_iu,  den = n_ac + n_au + n_ic + n_iu
    for (int k = 0; k < 21; ++k) {
        float th = umin + ((float)k / 20.0f) * du;
        float num = 0.0f, den = 0.0f;
        for (int r = tid; r < n; r += 1024) {
            float u = unc[r];
            float e = u / ENTMAX;
            float ratio = (e * (1.0f - 0.1f)) / fmaxf((1.0f - e) * 0.1f, EPS10);
            float v = 100.0f * __logf(fmaxf(ratio, EPS10));
            float s = 1.0f / (1.0f + __expf(-v));
            float t = tanhf(u);
            float cert = (1.0f - s) * (1.0f - t);
            float uw   = s * t;
            bool acc = (pred[r] == lbl0);
            bool le  = (u <= th);
            den += le ? cert : uw;
            num += le ? (acc ? cert : 0.0f) : (acc ? 0.0f : uw);
        }
        num = block_sum(num, red, wid, lane);
        den = block_sum(den, red, wid, lane);
        if (tid == 0) avu_s[k] = num / (den + EPS10);
        __syncthreads();
    }

    // Phase C: trapezoidal AUC + final scalars
    if (tid == 0) {
        float auc = 0.0f;
        for (int k = 1; k < 21; ++k) {
            float thk  = (float)k / 20.0f;
            float thk1 = (float)(k - 1) / 20.0f;
            auc += 0.5f * (avu_s[k] + avu_s[k - 1]) * (thk - thk1);
        }
        float savu = -__logf(fmaxf(auc, EPS10)) + foc_mean;   // BETA = 1
        out[0] = savu;
        out[1] = ce_mean;
    }
}

// ---------------------------------------------------------------------------
extern "C" void kernel_launch(void* const* d_in, const int* in_sizes, int n_in,
                              void* d_out, int out_size, void* d_ws, size_t ws_size,
                              hipStream_t stream) {
    const float* probs = (const float*)d_in[0];
    const float* y     = (const float*)d_in[1];
    const float* w     = (const float*)d_in[2];
    const int n = in_sizes[0] / NCLS;          // 32768

    float* ws      = (float*)d_ws;
    float* unc     = ws;                       // n floats
    float* ce_row  = ws + (size_t)n;           // n floats
    float* foc_row = ws + (size_t)2 * n;       // n floats
    int*   pred    = (int*)(ws + (size_t)3 * n); // n ints
    int*   label0  = pred + n;                 // 1 int

    float* out = (float*)d_out;

    const int blocks = (n + 7) / 8;            // 8 rows (waves) per 256-thr block
    savu_pass1<<<blocks, 256, 0, stream>>>(probs, y, w, unc, ce_row, foc_row,
                                           pred, label0, n);
    savu_pass2<<<1, 1024, 0, stream>>>(unc, ce_row, foc_row, pred, label0, out, n);
}